// Loss_8220567404602
// MI455X (gfx1250) — compile-verified
//
#include <hip/hip_runtime.h>

// ---------------------------------------------------------------------------
// YOLO loss on gfx1250 (MI455X).
// Bandwidth-bound: 193 MB streamed @ 23.3 TB/s => ~8.3 us floor.
// Strategy: one streaming pass (float2 loads, rows are 8B-aligned 120B records),
// per-thread f32 accumulation, LDS block reduce 256->64, then a
// V_WMMA_F32_16X16X4_F32 against an all-ones B matrix to collapse 64 partials
// per wave (A-matrix = 2 VGPRs/lane across wave32 covers exactly 64 floats).
// Deterministic 2-kernel reduction (no float atomics) so graph replays match.
// ---------------------------------------------------------------------------

typedef __attribute__((ext_vector_type(2))) float v2f;
typedef __attribute__((ext_vector_type(8))) float v8f;

#define NB_IMG   16384
#define SGRID    7
#define NCLS     20
#define NCELLS   (NB_IMG * SGRID * SGRID)   // 802816
#define NBLOCKS  1024
#define NTHREADS 256

// Reduce 64 floats spread as 2 per lane over a full wave32 using the f32 WMMA.
// A (16x4 f32): lanes 0-15 hold A[m][0..1], lanes 16-31 hold A[m][2..3].
// B = ones => D[m][n] = sum_k A[m][k]. Column 0 of D lives in lane 0
// (rows 0-7, one per VGPR) and lane 16 (rows 8-15). Sum-of-8 + shfl(16) done.
// Valid result on lane 0. EXEC must be all-ones in the calling wave.
__device__ __forceinline__ float wave_wmma_reduce64(float a0, float a1) {
    v2f a; a.x = a0; a.y = a1;
    v2f b; b.x = 1.0f; b.y = 1.0f;
    v8f c = {};
    c = __builtin_amdgcn_wmma_f32_16x16x4_f32(
        /*neg_a=*/false, a, /*neg_b=*/false, b,
        /*c_mod=*/(short)0, c, /*reuse_a=*/false, /*reuse_b=*/false);
    float s = c[0] + c[1] + c[2] + c[3] + c[4] + c[5] + c[6] + c[7];
    s += __shfl(s, 16, 32);   // lane0 += lane16 (wave32)
    return s;
}

__global__ void __launch_bounds__(NTHREADS)
yolo_loss_partial(const float* __restrict__ pred,
                  const float* __restrict__ targ,
                  float* __restrict__ partials) {
    const int tid = threadIdx.x;
    const float invS = 1.0f / (float)SGRID;
    float acc = 0.0f;

    for (int cell = blockIdx.x * NTHREADS + tid; cell < NCELLS;
         cell += NBLOCKS * NTHREADS) {
        const v2f* P2 = (const v2f*)(pred + (size_t)cell * 30);  // 120B rec, 8B aligned
        const v2f* T2 = (const v2f*)(targ + (size_t)cell * 30);

        float p[30], t[30];
#pragma unroll
        for (int i = 0; i < 15; ++i) { v2f v = P2[i]; p[2*i] = v.x; p[2*i+1] = v.y; }
#pragma unroll
        for (int i = 0; i < 15; ++i) { v2f v = T2[i]; t[2*i] = v.x; t[2*i+1] = v.y; }

        const int rem = cell % (SGRID * SGRID);
        const float fgx = (float)(rem % SGRID);   // meshgrid 'xy': add col to x
        const float fgy = (float)(rem / SGRID);   //               add row to y

        // Target box (slot 0 of target record).
        const float tw = t[2], th = t[3];
        const float gcx = (t[0] + fgx) * invS;
        const float gcy = (t[1] + fgy) * invS;
        const float g1x = gcx - 0.5f * tw, g2x = gcx + 0.5f * tw;
        const float g1y = gcy - 0.5f * th, g2y = gcy + 0.5f * th;
        const float area_g = tw * th;             // > 0 always (wh in [0.05,0.95])

        float iou[2], conf[2];
#pragma unroll
        for (int b = 0; b < 2; ++b) {
            const float pw = p[5*b+2], ph = p[5*b+3];
            conf[b] = p[5*b+4];
            const float pcx = (p[5*b+0] + fgx) * invS;
            const float pcy = (p[5*b+1] + fgy) * invS;
            const float p1x = pcx - 0.5f * pw, p2x = pcx + 0.5f * pw;
            const float p1y = pcy - 0.5f * ph, p2y = pcy + 0.5f * ph;
            float wx = fminf(g2x, p2x) - fmaxf(g1x, p1x);
            float wy = fminf(g2y, p2y) - fmaxf(g1y, p1y);
            wx = fmaxf(wx, 0.0f);
            wy = fmaxf(wy, 0.0f);
            const float inter  = wx * wy;
            const float area_p = pw * ph;
            iou[b] = inter / (area_g + area_p - inter);  // denom >= area_g > 0
        }

        const int   best     = (iou[1] > iou[0]) ? 1 : 0;  // argmax, first-on-tie
        const float best_iou = fmaxf(iou[0], iou[1]);

        float cl;
        if (t[4] == 1.0f) {                       // obj cell (obj is exactly 0/1)
            const float bx = p[5*best+0], by = p[5*best+1];
            const float bw = p[5*best+2], bh = p[5*best+3];
            const float dx = bx - t[0], dy = by - t[1];
            cl = 5.0f * (dx*dx + dy*dy);          // LAMBDA_COORD * xy

            const float pws = fminf(fmaxf(bw, 0.0f), 1.0f);
            const float phs = fminf(fmaxf(bh, 0.0f), 1.0f);
            const float sw = sqrtf(pws) - sqrtf(tw);
            const float sh = sqrtf(phs) - sqrtf(th);
            cl += 5.0f * (sw*sw + sh*sh);         // LAMBDA_COORD * wh

            const float dc = conf[best] - best_iou;
            cl += dc * dc;                        // positive conf
#pragma unroll
            for (int k = 0; k < NCLS; ++k) {      // class loss
                const float d = p[10 + k] - t[10 + k];
                cl += d * d;
            }
            const float oc = conf[1 - best];      // non-best box: neg_w = 1
            cl += 0.5f * oc * oc;                 // LAMBDA_NOOBJ
        } else {                                  // no-obj: both boxes neg_w = 1
            cl = 0.5f * (conf[0]*conf[0] + conf[1]*conf[1]);
        }
        acc += cl;
    }

    // Block reduction 256 -> 64 in LDS, then 64 -> 1 via WMMA on wave 0.
    __shared__ float smem[NTHREADS];
    smem[tid] = acc;
    __syncthreads();
    if (tid < 128) smem[tid] += smem[tid + 128];
    __syncthreads();
    if (tid < 64)  smem[tid] += smem[tid + 64];
    __syncthreads();
    if (tid < 32) {                               // wave 0, EXEC all-ones inside
        const float s = wave_wmma_reduce64(smem[tid], smem[tid + 32]);
        if (tid == 0) partials[blockIdx.x] = s;
    }
}

__global__ void __launch_bounds__(32)
yolo_loss_final(const float* __restrict__ partials, float* __restrict__ out) {
    const int lane = threadIdx.x;                 // single wave32
    float a0 = 0.0f, a1 = 0.0f;
    for (int i = lane; i < NBLOCKS; i += 64)        a0 += partials[i];
    for (int i = lane + 32; i < NBLOCKS; i += 64)   a1 += partials[i];
    const float s = wave_wmma_reduce64(a0, a1);
    if (lane == 0) {
        const float inv = 1.0f / (float)NB_IMG;   // exact power-of-two
        out[0] = s * inv * inv;                   // total / bs / bs
    }
}

extern "C" void kernel_launch(void* const* d_in, const int* in_sizes, int n_in,
                              void* d_out, int out_size, void* d_ws, size_t ws_size,
                              hipStream_t stream) {
    const float* pred = (const float*)d_in[0];    // (16384,7,7,30) f32
    const float* targ = (const float*)d_in[1];    // (16384,7,7,30) f32
    float* out      = (float*)d_out;              // scalar f32
    float* partials = (float*)d_ws;               // 1024 floats of scratch

    yolo_loss_partial<<<NBLOCKS, NTHREADS, 0, stream>>>(pred, targ, partials);
    yolo_loss_final<<<1, 32, 0, stream>>>(partials, out);
}